// GraphGuidedAttention_90486370992254
// MI455X (gfx1250) — compile-verified
//
#include <hip/hip_runtime.h>

#define B_  16
#define N_  1025
#define D_  512
#define H_  8
#define HD_ 64
#define QT_ 65          // ceil(1025/16) query tiles
#define ROWS_ 16400     // B_*N_ ; == 1025 row-tiles of 16 exactly

typedef __attribute__((ext_vector_type(2))) float v2f;
typedef __attribute__((ext_vector_type(8))) float v8f;

__device__ __forceinline__ v8f wmma_f32(v2f a, v2f b, v8f c) {
  // V_WMMA_F32_16X16X4_F32 : D = A(16x4) * B(4x16) + C(16x16)
  return __builtin_amdgcn_wmma_f32_16x16x4_f32(false, a, false, b, (short)0, c,
                                               false, false);
}

// ---------------------------------------------------------------- softplus
__global__ __launch_bounds__(256) void softplus_kernel(
    const float* __restrict__ gw, float* __restrict__ sp, int n) {
  int i = blockIdx.x * 256 + threadIdx.x;
  if (i < n) {
    float w = gw[i];
    sp[i] = (w > 20.0f) ? w : log1pf(__expf(w));
  }
}

// ---------------------------------------------------------------- QKV GEMM
// x [ROWS_, D_] @ w_qkv^T ([3D_, D_] row-major) -> scatter into Q/K/V
// [B,H,N,HD]. 16x64 strip per wave; software-pipelined loads (depth 1).
__global__ __launch_bounds__(32) void qkv_gemm(
    const float* __restrict__ x, const float* __restrict__ w,
    float* __restrict__ Qd, float* __restrict__ Kd, float* __restrict__ Vd) {
  const int lane = threadIdx.x;
  const int lm = lane & 15;
  const int lh = lane >> 4;          // selects K pair {0,1} vs {2,3}
  const int col0 = blockIdx.x * 64;  // 0..1472
  const int row0 = blockIdx.y * 16;  // 0..16384
  const float* ap = x + (size_t)(row0 + lm) * D_ + 2 * lh;
  const float* bp0 = w + (size_t)(col0 + 0 + lm) * D_ + 2 * lh;
  const float* bp1 = w + (size_t)(col0 + 16 + lm) * D_ + 2 * lh;
  const float* bp2 = w + (size_t)(col0 + 32 + lm) * D_ + 2 * lh;
  const float* bp3 = w + (size_t)(col0 + 48 + lm) * D_ + 2 * lh;
  v8f c0 = {}, c1 = {}, c2 = {}, c3 = {};
  // pipeline stage: fragments for current K step live one iteration ahead
  v2f a_c = *(const v2f*)(ap);
  v2f b0_c = *(const v2f*)(bp0);
  v2f b1_c = *(const v2f*)(bp1);
  v2f b2_c = *(const v2f*)(bp2);
  v2f b3_c = *(const v2f*)(bp3);
#pragma unroll 4
  for (int k = 0; k < D_ - 4; k += 4) {
    v2f a_n = *(const v2f*)(ap + k + 4);
    v2f b0_n = *(const v2f*)(bp0 + k + 4);
    v2f b1_n = *(const v2f*)(bp1 + k + 4);
    v2f b2_n = *(const v2f*)(bp2 + k + 4);
    v2f b3_n = *(const v2f*)(bp3 + k + 4);
    c0 = wmma_f32(a_c, b0_c, c0);
    c1 = wmma_f32(a_c, b1_c, c1);
    c2 = wmma_f32(a_c, b2_c, c2);
    c3 = wmma_f32(a_c, b3_c, c3);
    a_c = a_n; b0_c = b0_n; b1_c = b1_n; b2_c = b2_n; b3_c = b3_n;
  }
  c0 = wmma_f32(a_c, b0_c, c0);
  c1 = wmma_f32(a_c, b1_c, c1);
  c2 = wmma_f32(a_c, b2_c, c2);
  c3 = wmma_f32(a_c, b3_c, c3);

  // D layout: col = lane&15 (fixed per lane), row = i + 8*lh
  int bb8[8], nn8[8];
#pragma unroll
  for (int i = 0; i < 8; ++i) {
    int r = row0 + i + 8 * lh;  // global row = b*N_ + n
    int bbv = r / N_;
    bb8[i] = bbv;
    nn8[i] = r - bbv * N_;
  }
#pragma unroll
  for (int j = 0; j < 4; ++j) {
    const v8f& acc = (j == 0) ? c0 : (j == 1) ? c1 : (j == 2) ? c2 : c3;
    const int c = col0 + j * 16 + lm;
    const int part = c >> 9;  // 0=q 1=k 2=v
    const int rem = c & 511;
    const int h = rem >> 6;
    const int hd = rem & 63;
    float* dst = (part == 0) ? Qd : (part == 1) ? Kd : Vd;
#pragma unroll
    for (int i = 0; i < 8; ++i) {
      dst[(((size_t)(bb8[i] * H_ + h) * N_ + nn8[i]) * HD_) + hd] = acc[i];
    }
  }
}

// ---------------------------------------------------------------- attention
// One wave per (qtile, b*H+h). Flash-style streaming softmax over key tiles.
__global__ __launch_bounds__(32) void attn_kernel(
    const float* __restrict__ Qd, const float* __restrict__ Kd,
    const float* __restrict__ Vd, const float* __restrict__ SP,
    float* __restrict__ AO) {
  __shared__ float lds[16][18];  // P tile; stride 18 -> aligned v2f reads,
                                 // conflict-free (16 distinct even offsets)
  const int lane = threadIdx.x;
  const int lm = lane & 15;
  const int lh = lane >> 4;
  const int qt = blockIdx.x;  // 0..64
  const int bh = blockIdx.y;  // b*H + h
  const int b = bh >> 3;
  const int h = bh & 7;
  const size_t base = (size_t)bh * N_ * HD_;
  const float* Qp = Qd + base;
  const float* Kp = Kd + base;
  const float* Vp = Vd + base;
  const float scale = 0.125f;  // 64^-0.5

  const int qlane = qt * 16 + lm;  // A-matrix row for this lane
  const int qclamp = (qlane < N_) ? qlane : N_ - 1;
  v2f qa[16];
#pragma unroll
  for (int s = 0; s < 16; ++s)
    qa[s] = *(const v2f*)(Qp + (size_t)qclamp * HD_ + 4 * s + 2 * lh);

  v8f O0 = {}, O1 = {}, O2 = {}, O3 = {};
  float mrun[8], lrun[8];
#pragma unroll
  for (int i = 0; i < 8; ++i) { mrun[i] = -3.0e38f; lrun[i] = 0.0f; }

  for (int kt = 0; kt < QT_; ++kt) {
    // ---- S = Q @ K^T : preload all 16 B fragments, then chain WMMAs
    const int keyb = kt * 16 + lm;  // B-matrix col
    const int keybc = (keyb < N_) ? keyb : N_ - 1;
    const float* kp = Kp + (size_t)keybc * HD_ + 2 * lh;
    v2f bf[16];
#pragma unroll
    for (int s = 0; s < 16; ++s) bf[s] = *(const v2f*)(kp + 4 * s);
    v8f S = {};
#pragma unroll
    for (int s = 0; s < 16; ++s) S = wmma_f32(qa[s], bf[s], S);

    // ---- logits + streaming softmax update (rows live in 16-lane halves)
    const int keyd = kt * 16 + lm;  // D-layout col
    float p[8], corr[8];
#pragma unroll
    for (int i = 0; i < 8; ++i) {
      int qrow = qt * 16 + i + 8 * lh;
      float lg;
      if (qrow >= N_)      lg = 0.0f;      // dead row, never written
      else if (keyd >= N_) lg = -3.0e38f;  // masked key column
      else                 lg = S[i] * scale * SP[(size_t)qrow * N_ + keyd];
      float mx = lg;
#pragma unroll
      for (int off = 1; off < 16; off <<= 1)
        mx = fmaxf(mx, __shfl_xor(mx, off, 32));
      float mnew = fmaxf(mrun[i], mx);
      float pv = __expf(lg - mnew);
      float sum = pv;
#pragma unroll
      for (int off = 1; off < 16; off <<= 1) sum += __shfl_xor(sum, off, 32);
      float c = __expf(mrun[i] - mnew);
      lrun[i] = lrun[i] * c + sum;
      mrun[i] = mnew;
      corr[i] = c;
      p[i] = pv;
    }
#pragma unroll
    for (int i = 0; i < 8; ++i) {
      O0[i] *= corr[i]; O1[i] *= corr[i]; O2[i] *= corr[i]; O3[i] *= corr[i];
    }
    // ---- preload V fragments for this key tile (32 x b32, one clause)
    v2f vb[4][4];  // [kk][nt]
#pragma unroll
    for (int kk = 0; kk < 4; ++kk) {
      int krow = kt * 16 + 4 * kk + 2 * lh;
      int kr0 = (krow < N_) ? krow : N_ - 1;  // masked rows have P==0
      int kr1 = (krow + 1 < N_) ? krow + 1 : N_ - 1;
      const float* v0 = Vp + (size_t)kr0 * HD_;
      const float* v1 = Vp + (size_t)kr1 * HD_;
#pragma unroll
      for (int nt = 0; nt < 4; ++nt) {
        vb[kk][nt].x = v0[nt * 16 + lm];
        vb[kk][nt].y = v1[nt * 16 + lm];
      }
    }
    // ---- stage P through LDS to convert D-layout -> A-layout
    __syncthreads();
#pragma unroll
    for (int i = 0; i < 8; ++i) lds[i + 8 * lh][lm] = p[i];
    __syncthreads();
    // ---- O += P @ V   (K-dim = 16 keys -> 4 substeps; 4 hd subtiles)
#pragma unroll
    for (int kk = 0; kk < 4; ++kk) {
      v2f pa = *(const v2f*)&lds[lm][4 * kk + 2 * lh];
      O0 = wmma_f32(pa, vb[kk][0], O0);
      O1 = wmma_f32(pa, vb[kk][1], O1);
      O2 = wmma_f32(pa, vb[kk][2], O2);
      O3 = wmma_f32(pa, vb[kk][3], O3);
    }
  }
  // ---- finalize: AO[b, n, h*64 + hd]  (== transpose(0,2,1,3) reshape)
  float* outp = AO + (size_t)b * N_ * D_ + h * HD_;
#pragma unroll
  for (int i = 0; i < 8; ++i) {
    int qrow = qt * 16 + i + 8 * lh;
    if (qrow < N_) {
      float inv = 1.0f / lrun[i];
      size_t ro = (size_t)qrow * D_;
      outp[ro +  0 + lm] = O0[i] * inv;
      outp[ro + 16 + lm] = O1[i] * inv;
      outp[ro + 32 + lm] = O2[i] * inv;
      outp[ro + 48 + lm] = O3[i] * inv;
    }
  }
}

// ---------------------------------------------------------------- proj GEMM
// AO [ROWS_, D_] @ w_proj^T + b_proj -> P.  16x64 strip, pipelined loads.
__global__ __launch_bounds__(32) void proj_gemm(
    const float* __restrict__ AO, const float* __restrict__ w,
    const float* __restrict__ bias, float* __restrict__ P) {
  const int lane = threadIdx.x;
  const int lm = lane & 15;
  const int lh = lane >> 4;
  const int col0 = blockIdx.x * 64;  // 0..448
  const int row0 = blockIdx.y * 16;
  const float* ap = AO + (size_t)(row0 + lm) * D_ + 2 * lh;
  const float* bp0 = w + (size_t)(col0 + 0 + lm) * D_ + 2 * lh;
  const float* bp1 = w + (size_t)(col0 + 16 + lm) * D_ + 2 * lh;
  const float* bp2 = w + (size_t)(col0 + 32 + lm) * D_ + 2 * lh;
  const float* bp3 = w + (size_t)(col0 + 48 + lm) * D_ + 2 * lh;
  v8f c0 = {}, c1 = {}, c2 = {}, c3 = {};
  v2f a_c = *(const v2f*)(ap);
  v2f b0_c = *(const v2f*)(bp0);
  v2f b1_c = *(const v2f*)(bp1);
  v2f b2_c = *(const v2f*)(bp2);
  v2f b3_c = *(const v2f*)(bp3);
#pragma unroll 4
  for (int k = 0; k < D_ - 4; k += 4) {
    v2f a_n = *(const v2f*)(ap + k + 4);
    v2f b0_n = *(const v2f*)(bp0 + k + 4);
    v2f b1_n = *(const v2f*)(bp1 + k + 4);
    v2f b2_n = *(const v2f*)(bp2 + k + 4);
    v2f b3_n = *(const v2f*)(bp3 + k + 4);
    c0 = wmma_f32(a_c, b0_c, c0);
    c1 = wmma_f32(a_c, b1_c, c1);
    c2 = wmma_f32(a_c, b2_c, c2);
    c3 = wmma_f32(a_c, b3_c, c3);
    a_c = a_n; b0_c = b0_n; b1_c = b1_n; b2_c = b2_n; b3_c = b3_n;
  }
  c0 = wmma_f32(a_c, b0_c, c0);
  c1 = wmma_f32(a_c, b1_c, c1);
  c2 = wmma_f32(a_c, b2_c, c2);
  c3 = wmma_f32(a_c, b3_c, c3);
#pragma unroll
  for (int j = 0; j < 4; ++j) {
    const v8f& acc = (j == 0) ? c0 : (j == 1) ? c1 : (j == 2) ? c2 : c3;
    const int c = col0 + j * 16 + lm;
    const float bc = bias[c];
#pragma unroll
    for (int i = 0; i < 8; ++i) {
      int r = row0 + i + 8 * lh;
      P[(size_t)r * D_ + c] = acc[i] + bc;
    }
  }
}

// ------------------------------------------------------- LayerNorm+residual
// In-place on d_out: y <- y + LN(y)*gamma + beta   (one block per row)
__global__ __launch_bounds__(256) void ln_residual(
    float* __restrict__ Y, const float* __restrict__ gamma,
    const float* __restrict__ beta) {
  __shared__ float reds[8], reds2[8];
  __shared__ float mu_s, rs_s;
  const int row = blockIdx.x;
  float* y = Y + (size_t)row * D_;
  const int t = threadIdx.x;
  float a = y[t], bv = y[t + 256];
  float s = a + bv, s2 = a * a + bv * bv;
#pragma unroll
  for (int off = 1; off < 32; off <<= 1) {
    s += __shfl_xor(s, off, 32);
    s2 += __shfl_xor(s2, off, 32);
  }
  if ((t & 31) == 0) { reds[t >> 5] = s; reds2[t >> 5] = s2; }
  __syncthreads();
  if (t == 0) {
    float S = 0.f, S2 = 0.f;
#pragma unroll
    for (int i = 0; i < 8; ++i) { S += reds[i]; S2 += reds2[i]; }
    float mu = S * (1.0f / D_);
    float var = S2 * (1.0f / D_) - mu * mu;
    mu_s = mu;
    rs_s = rsqrtf(var + 1e-5f);
  }
  __syncthreads();
  const float mu = mu_s, rs = rs_s;
  y[t]       = a  + (a  - mu) * rs * gamma[t]       + beta[t];
  y[t + 256] = bv + (bv - mu) * rs * gamma[t + 256] + beta[t + 256];
}

// ---------------------------------------------------------------- launcher
extern "C" void kernel_launch(void* const* d_in, const int* in_sizes, int n_in,
                              void* d_out, int out_size, void* d_ws,
                              size_t ws_size, hipStream_t stream) {
  const float* x      = (const float*)d_in[0];  // [B,N,D]
  const float* w_qkv  = (const float*)d_in[1];  // [3D,D]
  const float* w_proj = (const float*)d_in[2];  // [D,D]
  const float* b_proj = (const float*)d_in[3];  // [D]
  const float* gw     = (const float*)d_in[4];  // [N,N]
  const float* ln_g   = (const float*)d_in[5];  // [D]
  const float* ln_b   = (const float*)d_in[6];  // [D]
  float* out = (float*)d_out;                   // [B,N,D]

  float* ws = (float*)d_ws;
  const size_t nsp  = (size_t)N_ * N_;             // 1,050,625
  const size_t nqkv = (size_t)B_ * H_ * N_ * HD_;  // 8,396,800
  float* SP = ws;
  float* Q  = SP + nsp;
  float* K  = Q + nqkv;
  float* V  = K + nqkv;
  float* AO = V + nqkv;

  softplus_kernel<<<(int)((nsp + 255) / 256), 256, 0, stream>>>(gw, SP,
                                                                (int)nsp);
  qkv_gemm<<<dim3(3 * D_ / 64, ROWS_ / 16), 32, 0, stream>>>(x, w_qkv, Q, K,
                                                             V);
  attn_kernel<<<dim3(QT_, B_ * H_), 32, 0, stream>>>(Q, K, V, SP, AO);
  proj_gemm<<<dim3(D_ / 64, ROWS_ / 16), 32, 0, stream>>>(AO, w_proj, b_proj,
                                                          out);
  ln_residual<<<ROWS_, 256, 0, stream>>>(out, ln_g, ln_b);
}